// LossCompute_12378095747451
// MI455X (gfx1250) — compile-verified
//
#include <hip/hip_runtime.h>

#define P_CONST 5.0f
#define A_CONST 10.0f

typedef __attribute__((ext_vector_type(2))) float v2f;
typedef __attribute__((ext_vector_type(8))) float v8f;

__global__ void zero_ws_kernel(float* __restrict__ p, int n) {
    int i = blockIdx.x * blockDim.x + threadIdx.x;
    if (i < n) p[i] = 0.0f;
}

// One thread processes 4 edges of one polarity.
// int64 index rows are read as int4 (B128) chunks; only low dwords are used
// since all indices fit in 32 bits.
__global__ __launch_bounds__(256) void edge_kernel(
    const float* __restrict__ x,
    const long long* __restrict__ adj_pos,
    const long long* __restrict__ adj_neg,
    float* __restrict__ num,
    float* __restrict__ dom,
    int E, int tpp /* threads per polarity */)
{
    int t = blockIdx.x * blockDim.x + threadIdx.x;
    int pol = (t >= tpp) ? 1 : 0;
    int base = (t - pol * tpp) * 4;
    if (base >= E || t >= 2 * tpp) return;

    const long long* adj = pol ? adj_neg : adj_pos;

    if (base + 3 < E) {
        const int4* crow = (const int4*)adj;        // clause row (row 0)
        const int4* vrow = (const int4*)(adj + E);  // var row    (row 1)
        int4 c01 = crow[(base >> 1) + 0];
        int4 c23 = crow[(base >> 1) + 1];
        int4 v01 = vrow[(base >> 1) + 0];
        int4 v23 = vrow[(base >> 1) + 1];
        int cs[4] = {c01.x, c01.z, c23.x, c23.z};
        int vs[4] = {v01.x, v01.z, v23.x, v23.z};
#pragma unroll
        for (int j = 0; j < 4; ++j) {
            float tv = x[vs[j]];
            if (pol) tv = 1.0f - tv;
            float w = __expf(P_CONST * tv);
            atomicAdd(&num[cs[j]], tv * w);   // non-returning global_atomic_add_f32
            atomicAdd(&dom[cs[j]], w);
        }
    } else {
        for (int j = base; j < E; ++j) {
            int ci = (int)adj[j];
            int vi = (int)adj[E + j];
            float tv = x[vi];
            if (pol) tv = 1.0f - tv;
            float w = __expf(P_CONST * tv);
            atomicAdd(&num[ci], tv * w);
            atomicAdd(&dom[ci], w);
        }
    }
}

// Grid-stride squared-error accumulation, then wave reduction via
// V_WMMA_F32_16X16X4_F32 with B = ones:
//   D[m][n] = sum_k A[m][k]  ->  row m = partial[lane m] + partial[lane m+16]
//   rows 0..7  -> lane 0's 8 D components, rows 8..15 -> lane 16's.
__global__ __launch_bounds__(256) void clause_kernel(
    const float* __restrict__ num,
    const float* __restrict__ dom,
    const float* __restrict__ cc,
    float* __restrict__ accum,
    int C)
{
    int gid    = blockIdx.x * blockDim.x + threadIdx.x;
    int stride = gridDim.x * blockDim.x;

    float partial = 0.0f;
    for (int i = gid; i < C; i += stride) {
        float r  = num[i] / dom[i];
        float tt = A_CONST * (r - 0.5f);
        float sm = 1.0f / (1.0f + __expf(-tt));
        float df = sm - cc[i];
        partial += df * df;
    }

    // EXEC is all-1s here (uniform control flow, full 256-thread blocks).
    v2f a; a[0] = partial; a[1] = 0.0f;
    v2f b; b[0] = 1.0f;    b[1] = 1.0f;
    v8f c = {};
    v8f d = __builtin_amdgcn_wmma_f32_16x16x4_f32(
        /*neg_a=*/false, a, /*neg_b=*/false, b,
        /*c_mod=*/(short)0, c, /*reuse_a=*/false, /*reuse_b=*/false);

    float s = ((d[0] + d[1]) + (d[2] + d[3])) + ((d[4] + d[5]) + (d[6] + d[7]));

    __shared__ float red[16];   // 8 waves x 2 half-sums
    int lane = threadIdx.x & 31;
    int wave = threadIdx.x >> 5;
    if ((lane & 15) == 0) red[wave * 2 + (lane >> 4)] = s;
    __syncthreads();

    if (threadIdx.x == 0) {
        float tot = 0.0f;
#pragma unroll
        for (int i = 0; i < 16; ++i) tot += red[i];
        atomicAdd(accum, tot);
    }
}

__global__ void finalize_kernel(const float* __restrict__ accum,
                                float* __restrict__ out, float invC) {
    out[0] = accum[0] * invC;
}

extern "C" void kernel_launch(void* const* d_in, const int* in_sizes, int n_in,
                              void* d_out, int out_size, void* d_ws, size_t ws_size,
                              hipStream_t stream) {
    const float*     x       = (const float*)d_in[0];      // xv [V]
    const long long* adj_pos = (const long long*)d_in[1];  // [2, E] int64
    const long long* adj_neg = (const long long*)d_in[2];  // [2, E] int64
    const float*     cc      = (const float*)d_in[3];      // clause_count [C]

    int E = in_sizes[1] / 2;
    int C = in_sizes[3];

    float* num   = (float*)d_ws;   // [C]
    float* dom   = num + C;        // [C]
    float* accum = dom + C;        // [1]

    int zn = 2 * C + 1;
    zero_ws_kernel<<<(zn + 255) / 256, 256, 0, stream>>>(num, zn);

    int tpp = (E + 3) / 4;                 // threads per polarity
    int tot = 2 * tpp;
    edge_kernel<<<(tot + 255) / 256, 256, 0, stream>>>(x, adj_pos, adj_neg,
                                                       num, dom, E, tpp);

    int cblocks = (C + 255) / 256;
    if (cblocks > 1024) cblocks = 1024;
    clause_kernel<<<cblocks, 256, 0, stream>>>(num, dom, cc, accum, C);

    finalize_kernel<<<1, 1, 0, stream>>>(accum, (float*)d_out, 1.0f / (float)C);
}